// FLoss_19774029431480
// MI455X (gfx1250) — compile-verified
//
#include <hip/hip_runtime.h>
#include <hip/hip_bf16.h>
#include <math.h>

// ---- problem constants (reference: B=16, T=16, W=256) ----
#define WDIM        256
#define FRAMES      256                       // B*T
#define FRAME_ELEMS (WDIM * WDIM)             // 65536
#define N_TOTAL     ((float)FRAMES * (float)FRAME_ELEMS)
#define BLOCKS_PER_FRAME 8
#define ROWS_PER_BLOCK   (WDIM / BLOCKS_PER_FRAME)   // 32
#define NPART       (FRAMES * BLOCKS_PER_FRAME)      // 2048

typedef __attribute__((ext_vector_type(2))) float v2f;
typedef __attribute__((ext_vector_type(8))) float v8f;

// ---------------------------------------------------------------------------
// Kernel 1: per-frame argmax centroid.  One 256-thread block per frame.
// Pass 1: max over 256x256 frame (column-per-thread, coalesced rows).
// Pass 2: re-scan (L2-resident) counting argmax pixels and their row/col sums.
// Writes fparam[2f] = mean row (x), fparam[2f+1] = mean col (y).
// ---------------------------------------------------------------------------
__global__ __launch_bounds__(256) void centroid_kernel(
    const float* __restrict__ tgt, float* __restrict__ fparam) {
  const int f = blockIdx.x;
  const float* base = tgt + (size_t)f * FRAME_ELEMS;
  const int t = threadIdx.x;  // column index

  __shared__ float red[256];
  __shared__ float rc[256], rx[256], ry[256];

  // pass 1: frame max
  float m = -INFINITY;
  for (int r = 0; r < WDIM; ++r) m = fmaxf(m, base[r * WDIM + t]);
  red[t] = m;
  __syncthreads();
  for (int s = 128; s > 0; s >>= 1) {
    if (t < s) red[t] = fmaxf(red[t], red[t + s]);
    __syncthreads();
  }
  m = red[0];
  __syncthreads();

  // pass 2: centroid of maximal pixels (exact float equality, max is exact)
  float cnt = 0.0f, sx = 0.0f;
  for (int r = 0; r < WDIM; ++r) {
    if (base[r * WDIM + t] == m) { cnt += 1.0f; sx += (float)r; }
  }
  rc[t] = cnt;
  rx[t] = sx;
  ry[t] = cnt * (float)t;  // this thread owns column t
  __syncthreads();
  for (int s = 128; s > 0; s >>= 1) {
    if (t < s) { rc[t] += rc[t + s]; rx[t] += rx[t + s]; ry[t] += ry[t + s]; }
    __syncthreads();
  }
  if (t == 0) {
    const float inv = 1.0f / rc[0];
    fparam[2 * f + 0] = rx[0] * inv;  // x = mean row index
    fparam[2 * f + 1] = ry[0] * inv;  // y = mean col index
  }
}

// ---------------------------------------------------------------------------
// Kernel 2: weighted-BCE partial sums.  grid = (8 chunks, 256 frames),
// 256 threads/block, float4 loads; fixed-order LDS tree -> deterministic.
// ---------------------------------------------------------------------------
__device__ __forceinline__ float bce_term(float p, float tv, float d2) {
  const float dist = sqrtf(d2);
  const float w = 256.0f / (dist + 1.0f);
  const float lp  = fmaxf(logf(p),     -100.0f);  // log(0) = -inf -> -100
  const float l1p = fmaxf(log1pf(-p),  -100.0f);
  return w * (tv * lp + (1.0f - tv) * l1p);
}

__global__ __launch_bounds__(256) void loss_partial_kernel(
    const float* __restrict__ inp, const float* __restrict__ tgt,
    const float* __restrict__ fparam, float* __restrict__ part) {
  const int f     = blockIdx.y;   // frame
  const int chunk = blockIdx.x;   // row chunk within frame
  const float x = fparam[2 * f + 0];
  const float y = fparam[2 * f + 1];

  const size_t fbase = (size_t)f * FRAME_ELEMS;
  const float* P = inp + fbase;
  const float* T = tgt + fbase;

  const int t    = threadIdx.x;
  const int j0   = (t & 63) * 4;  // starting column of this thread's float4
  const int rsub = t >> 6;        // 0..3: row within a 4-row group
  const int row0 = chunk * ROWS_PER_BLOCK;

  float dy2[4];
#pragma unroll
  for (int k = 0; k < 4; ++k) {
    const float dy = (float)(j0 + k) - y;
    dy2[k] = dy * dy;
  }

  float acc = 0.0f;
#pragma unroll
  for (int rg = 0; rg < ROWS_PER_BLOCK; rg += 4) {
    const int row  = row0 + rg + rsub;
    const float dx = (float)row - x;
    const float dx2 = dx * dx;
    const float4 p4 = *(const float4*)(P + (size_t)row * WDIM + j0);
    const float4 t4 = *(const float4*)(T + (size_t)row * WDIM + j0);
    acc += bce_term(p4.x, t4.x, dx2 + dy2[0]);
    acc += bce_term(p4.y, t4.y, dx2 + dy2[1]);
    acc += bce_term(p4.z, t4.z, dx2 + dy2[2]);
    acc += bce_term(p4.w, t4.w, dx2 + dy2[3]);
  }

  __shared__ float red[256];
  red[t] = acc;
  __syncthreads();
  for (int s = 128; s > 0; s >>= 1) {
    if (t < s) red[t] += red[t + s];
    __syncthreads();
  }
  if (t == 0) part[f * BLOCKS_PER_FRAME + chunk] = red[0];
}

// ---------------------------------------------------------------------------
// Kernel 3: final reduction of 2048 partials with V_WMMA_F32_16X16X4_F32.
// One wave (32 lanes, EXEC all-ones, no divergence before WMMA).
// A = 64 partials per step (16x4 f32, 2 VGPRs/lane), B = ones (4x16),
// C accumulates: D[m,n] = C[m,n] + sum_k A[m,k].  Every column of D holds the
// row sums, so sum(D) = 16 * total.  Exact /16, deterministic shfl tree.
// ---------------------------------------------------------------------------
__global__ __launch_bounds__(32) void final_reduce_kernel(
    const float* __restrict__ part, float* __restrict__ out) {
  const int l = threadIdx.x;

  v8f c = {0.f, 0.f, 0.f, 0.f, 0.f, 0.f, 0.f, 0.f};
  v2f b;
  b.x = 1.0f;
  b.y = 1.0f;

  for (int base = 0; base < NPART; base += 64) {
    v2f a;
    a.x = part[base + l];
    a.y = part[base + 32 + l];
    // 8 args: (neg_a, A, neg_b, B, c_mod, C, reuse_a, reuse_b)
    c = __builtin_amdgcn_wmma_f32_16x16x4_f32(
        false, a, false, b, (short)0, c, false, false);
  }

  float s = c[0] + c[1] + c[2] + c[3] + c[4] + c[5] + c[6] + c[7];
#pragma unroll
  for (int off = 16; off > 0; off >>= 1) s += __shfl_xor(s, off, 32);

  if (l == 0) out[0] = -(s * (1.0f / 16.0f)) / N_TOTAL;
}

// ---------------------------------------------------------------------------
extern "C" void kernel_launch(void* const* d_in, const int* in_sizes, int n_in,
                              void* d_out, int out_size, void* d_ws, size_t ws_size,
                              hipStream_t stream) {
  const float* inp = (const float*)d_in[0];  // [16,1,16,256,256] f32 (flat == squeezed)
  const float* tgt = (const float*)d_in[1];  // [16,16,256,256] f32
  float* out = (float*)d_out;

  float* fparam = (float*)d_ws;              // 512 floats: (x,y) per frame
  float* part   = (float*)d_ws + 1024;       // 2048 floats: block partials

  centroid_kernel<<<FRAMES, 256, 0, stream>>>(tgt, fparam);
  loss_partial_kernel<<<dim3(BLOCKS_PER_FRAME, FRAMES), 256, 0, stream>>>(
      inp, tgt, fparam, part);
  final_reduce_kernel<<<1, 32, 0, stream>>>(part, out);
}